// FRFT_16492674417379
// MI455X (gfx1250) — compile-verified
//
#include <hip/hip_runtime.h>
#include <stdint.h>

typedef __attribute__((ext_vector_type(2))) float v2f;
typedef __attribute__((ext_vector_type(4))) float v4f;

// ---------------------------------------------------------------------------
// Kernel 1: build the 256-entry complex phase table into d_ws.
//   T[j] = 16 * exp(-i * (pi/2) * a * l[(j+128)%256])
// with l[m] = m except l[255] = 256 (the "even" fix-up; (j+128)%256==255 <=> j==127).
// One table serves both H and W since H == W == 256. Each axis carries a
// sqrt(256)=16 factor, so the product applied per element is 256 * dh * dw.
// Precise sinf/cosf kept deliberately: this kernel is 1 block / 256 threads,
// cost is negligible, and it matches the reference's complex64 exp closely.
// ---------------------------------------------------------------------------
__global__ void frft_table_k(const float* __restrict__ order,
                             float* __restrict__ tab) {
    int j = threadIdx.x;                 // 0..255
    float a = order[0];
    int   m  = (j + 128) & 255;
    float lv = (m == 255) ? 256.0f : (float)m;
    float th = -1.5707963267948966f * a * lv;
    v2f t;
    t.x = 16.0f * cosf(th);
    t.y = 16.0f * sinf(th);
    *(v2f*)(tab + 2 * j) = t;            // one b64 store
}

// ---------------------------------------------------------------------------
// Kernel 2: streaming elementwise complex scale (the einsums are multiplies
// by DIAGONAL matrices; the double fftshift on even axes cancels on x).
// Thread t handles x elements e=2t, e=2t+1 (same h row; rows are 256 wide and
// e is even), writes interleaved complex [re0,im0,re1,im1] as one float4.
// Input staged via CDNA5 async global->LDS load (ASYNCcnt), output via
// non-temporal b128 store (write-once 256MB stream; don't churn the 192MB L2).
// ---------------------------------------------------------------------------
__global__ void __launch_bounds__(256) frft_main_k(const float* __restrict__ x,
                                                   const float* __restrict__ tab,
                                                   float* __restrict__ out) {
    __shared__ float tile[512];          // 2 floats per thread

    int tid = threadIdx.x;
    long long t = (long long)blockIdx.x * 256 + tid;
    long long e = t * 2;                 // first x element for this thread
    int w = (int)(e & 255);              // even
    int h = (int)((e >> 8) & 255);

    // Async copy of this lane's 8 bytes of x into LDS, then wait on ASYNCcnt.
    // Per wave: one instruction moving 256B contiguous global -> LDS.
    unsigned long long gaddr = (unsigned long long)(uintptr_t)(x + e);
    unsigned int       laddr = (unsigned int)(uintptr_t)&tile[2 * tid];
    asm volatile("global_load_async_to_lds_b64 %0, %1, off\n\t"
                 "s_wait_asynccnt 0x0"
                 :: "v"(laddr), "v"(gaddr)
                 : "memory");
    float x0 = tile[2 * tid];
    float x1 = tile[2 * tid + 1];

    // Complex phase factors (2KB table, hot in cache).
    // tab + 2*w is 16B-aligned (w even) -> single b128 load for both w factors.
    v2f dh = *(const v2f*)(tab + 2 * h);
    v4f dw = *(const v4f*)(tab + 2 * w); // (d0.re, d0.im, d1.re, d1.im)

    // out = (dh * dw) * x   (complex * complex, then * real x), matching the
    // reference's per-axis application order (Fh applied, then Fw).
    v4f o;
    o.x = (dh.x * dw.x - dh.y * dw.y) * x0;
    o.y = (dh.x * dw.y + dh.y * dw.x) * x0;
    o.z = (dh.x * dw.z - dh.y * dw.w) * x1;
    o.w = (dh.x * dw.w + dh.y * dw.z) * x1;

    __builtin_nontemporal_store(o, (v4f*)(out + 2 * e));
}

// ---------------------------------------------------------------------------
extern "C" void kernel_launch(void* const* d_in, const int* in_sizes, int n_in,
                              void* d_out, int out_size, void* d_ws, size_t ws_size,
                              hipStream_t stream) {
    const float* x     = (const float*)d_in[0];   // (8,64,256,256) f32
    const float* order = (const float*)d_in[1];   // scalar f32
    float*       tab   = (float*)d_ws;            // 256 complex = 2KB scratch
    float*       out   = (float*)d_out;           // complex64 interleaved -> 2*N floats

    long long total   = (long long)in_sizes[0];   // 8*64*256*256 elements
    long long threads = total / 2;                // 2 elements per thread
    int blocks = (int)(threads / 256);            // 65536

    frft_table_k<<<1, 256, 0, stream>>>(order, tab);
    frft_main_k<<<blocks, 256, 0, stream>>>(x, tab, out);
}